// EnhancedDNCMemory_79783312490778
// MI455X (gfx1250) — compile-verified
//
#include <hip/hip_runtime.h>
#include <math.h>

// MI455X / gfx1250, wave32. Dominant cost: one streaming pass over link
// (268 MB ~= 11.5us at 23.3 TB/s). link_new is never materialized; its two
// einsums are refactored into L*X and L^T*X (X = 8 useful cols padded to 16)
// evaluated with v_wmma_f32_16x16x4_f32. L tiles are quad-buffered in LDS via
// GLOBAL_LOAD_ASYNC_TO_LDS with distance-2 prefetch (ASYNCcnt <= 3 drain), and
// the end-of-iteration barrier is hand-rolled so no s_wait_storecnt lands on
// the critical path.

typedef __attribute__((ext_vector_type(2))) float v2f;
typedef __attribute__((ext_vector_type(8))) float v8f;

#define NB 4
#define NN 4096
#define NW 64
#define NR 4
#define EPSF 1e-6f

__device__ __forceinline__ float sigm(float x) { return 1.f / (1.f + expf(-x)); }
__device__ __forceinline__ float softplus_(float x) {
  return fmaxf(x, 0.f) + log1pf(expf(-fabsf(x)));
}

// ---------------- workspace layout (floats) ----------------
constexpr size_t OFF_CW  = 0;                                   // B*N  write content weights
constexpr size_t OFF_WW  = OFF_CW + (size_t)NB * NN;            // B*N  write weights
constexpr size_t OFF_S   = OFF_WW + (size_t)NB * NN;            // B*R  <p, rwp_r>
constexpr size_t OFF_T   = OFF_S + (size_t)NB * NR;             // B*R  <w, rwp_r>
constexpr size_t OFF_X   = OFF_T + (size_t)NB * NR;             // B*N*16 X matrix (8 used cols)
constexpr size_t OFF_MN  = OFF_X + (size_t)NB * NN * 16;        // B*N*W memory_new
constexpr size_t OFF_NRM = OFF_MN + (size_t)NB * NN * NW;       // B*N  |memory_new| rows
constexpr size_t OFF_RA  = OFF_NRM + (size_t)NB * NN;           // B*N*16 row accum (L*X)
constexpr size_t OFF_PC  = OFF_RA + (size_t)NB * NN * 16;       // B*32*N*16 col partials
constexpr size_t OFF_CA  = OFF_PC + (size_t)NB * 32 * NN * 16;  // B*N*16 col accum (L^T*X)
constexpr size_t OFF_FWD = OFF_CA + (size_t)NB * NN * 16;       // B*R*N
constexpr size_t OFF_BWD = OFF_FWD + (size_t)NB * NR * NN;      // B*R*N
constexpr size_t OFF_DOT = OFF_BWD + (size_t)NB * NR * NN;      // B*R*N read dots
constexpr size_t OFF_RDW = OFF_DOT + (size_t)NB * NR * NN;      // B*R*N read weights

// ---------------- 1) write-key content weights (softmax over N) ----------------
__global__ void __launch_bounds__(1024) k_content_write(
    const float* __restrict__ mem, const float* __restrict__ wkey,
    const float* __restrict__ wsr, float* __restrict__ cw) {
  int b = blockIdx.x;
  int tid = threadIdx.x;
  __shared__ float simv[NN];
  __shared__ float red[1024];
  float kn2 = 0.f;
  for (int w = 0; w < NW; ++w) { float k = wkey[b * NW + w]; kn2 += k * k; }
  float keyn = sqrtf(kn2);
  float beta = 1.f + softplus_(wsr[b]);
  for (int p = 0; p < 4; ++p) {
    int n = tid + p * 1024;
    const float* row = mem + ((size_t)b * NN + n) * NW;
    float dot = 0.f, ss = 0.f;
    for (int w = 0; w < NW; w += 4) {
      float4 m = *(const float4*)(row + w);
      float4 k = *(const float4*)(wkey + b * NW + w);
      dot += m.x * k.x + m.y * k.y + m.z * k.z + m.w * k.w;
      ss += m.x * m.x + m.y * m.y + m.z * m.z + m.w * m.w;
    }
    simv[n] = beta * dot / (sqrtf(ss) * keyn + EPSF);
  }
  __syncthreads();
  float mx = -3.4e38f;
  for (int p = 0; p < 4; ++p) mx = fmaxf(mx, simv[tid + p * 1024]);
  red[tid] = mx;
  __syncthreads();
  for (int off = 512; off > 0; off >>= 1) {
    if (tid < off) red[tid] = fmaxf(red[tid], red[tid + off]);
    __syncthreads();
  }
  mx = red[0];
  __syncthreads();
  float ps = 0.f;
  for (int p = 0; p < 4; ++p) {
    int n = tid + p * 1024;
    float e = expf(simv[n] - mx);
    simv[n] = e;
    ps += e;
  }
  red[tid] = ps;
  __syncthreads();
  for (int off = 512; off > 0; off >>= 1) {
    if (tid < off) red[tid] += red[tid + off];
    __syncthreads();
  }
  float inv = 1.f / red[0];
  for (int p = 0; p < 4; ++p) {
    int n = tid + p * 1024;
    cw[b * NN + n] = simv[n] * inv;
  }
}

// ------- 2) usage/retention, bitonic-sort allocation, write weight, X matrix -------
__global__ void __launch_bounds__(1024) k_alloc(
    const float* __restrict__ usage, const float* __restrict__ rwp,
    const float* __restrict__ wwp, const float* __restrict__ prec,
    const float* __restrict__ fraw, const float* __restrict__ agr,
    const float* __restrict__ wgr, const float* __restrict__ cw,
    float* __restrict__ ww, float* __restrict__ xbuf,
    float* __restrict__ sv, float* __restrict__ tv) {
  int b = blockIdx.x;
  int tid = threadIdx.x;
  __shared__ float vals[NN];
  __shared__ int idxs[NN];
  __shared__ float allocA[NN];
  __shared__ float scanA[1024], scanB[1024];
  __shared__ float small[8];

  float fr[NR];
  for (int r = 0; r < NR; ++r) fr[r] = sigm(fraw[b * NR + r]);
  float ga = sigm(agr[b]);
  float gw = sigm(wgr[b]);

  for (int p = 0; p < 4; ++p) {
    int n = tid + p * 1024;
    float ret = 1.f;
    for (int r = 0; r < NR; ++r)
      ret *= (1.f - fr[r] * rwp[(size_t)(b * NR + r) * NN + n]);
    float u = usage[b * NN + n], wp = wwp[b * NN + n];
    vals[n] = (u + wp - u * wp) * ret;
    idxs[n] = n;
  }
  __syncthreads();

  // bitonic sort ascending, stable via index tie-break
  for (int k = 2; k <= NN; k <<= 1) {
    for (int j = k >> 1; j > 0; j >>= 1) {
      for (int p = 0; p < 4; ++p) {
        int i = tid + p * 1024;
        int ix = i ^ j;
        if (ix > i) {
          bool up = ((i & k) == 0);
          float a = vals[i], c = vals[ix];
          int ia = idxs[i], ic = idxs[ix];
          bool agtb = (a > c) || (a == c && ia > ic);
          if (agtb == up) { vals[i] = c; vals[ix] = a; idxs[i] = ic; idxs[ix] = ia; }
        }
      }
      __syncthreads();
    }
  }

  // exclusive multiplicative scan (chunk-of-4 + Hillis-Steele over 1024 chunks)
  float u0 = vals[tid * 4 + 0], u1 = vals[tid * 4 + 1];
  float u2 = vals[tid * 4 + 2], u3 = vals[tid * 4 + 3];
  float c01 = u0 * u1;
  scanA[tid] = c01 * u2 * u3;
  __syncthreads();
  float* src = scanA; float* dst = scanB;
  for (int off = 1; off < 1024; off <<= 1) {
    float v = src[tid];
    if (tid >= off) v *= src[tid - off];
    dst[tid] = v;
    __syncthreads();
    float* tmp = src; src = dst; dst = tmp;
  }
  float ex = (tid == 0) ? 1.f : src[tid - 1];
  allocA[idxs[tid * 4 + 0]] = (1.f - u0) * ex;
  allocA[idxs[tid * 4 + 1]] = (1.f - u1) * ex * u0;
  allocA[idxs[tid * 4 + 2]] = (1.f - u2) * ex * c01;
  allocA[idxs[tid * 4 + 3]] = (1.f - u3) * ex * c01 * u2;
  __syncthreads();

  // w_w, X matrix, and the scalar reductions s[r], t[r]
  float part[8];
  for (int q = 0; q < 8; ++q) part[q] = 0.f;
  for (int p = 0; p < 4; ++p) {
    int n = tid + p * 1024;
    float w = gw * (ga * allocA[n] + (1.f - ga) * cw[b * NN + n]);
    ww[b * NN + n] = w;
    float pn = prec[b * NN + n];
    size_t xo = ((size_t)(b * NN) + n) * 16;
    for (int r = 0; r < NR; ++r) {
      float x = rwp[(size_t)(b * NR + r) * NN + n];
      xbuf[xo + r] = x;
      xbuf[xo + 4 + r] = w * x;
      part[r] += pn * x;
      part[4 + r] += w * x;
    }
    for (int c = 8; c < 16; ++c) xbuf[xo + c] = 0.f;
  }
  for (int q = 0; q < 8; ++q) {
    __syncthreads();
    scanA[tid] = part[q];
    __syncthreads();
    for (int off = 512; off > 0; off >>= 1) {
      if (tid < off) scanA[tid] += scanA[tid + off];
      __syncthreads();
    }
    if (tid == 0) small[q] = scanA[0];
  }
  __syncthreads();
  if (tid < 4) sv[b * NR + tid] = small[tid];
  else if (tid < 8) tv[b * NR + tid - 4] = small[tid];
}

// ---------------- 3) memory_new + row norms ----------------
__global__ void __launch_bounds__(256) k_mem_update(
    const float* __restrict__ mem, const float* __restrict__ eraw,
    const float* __restrict__ wvec, const float* __restrict__ ww,
    float* __restrict__ mnew, float* __restrict__ mnorm) {
  int t = threadIdx.x, lane = t & 31, wv = t >> 5;
  int row = blockIdx.x * 8 + wv;
  int b = row >> 12, n = row & 4095;
  int w2 = lane * 2;
  float e0 = sigm(eraw[b * NW + w2]), e1 = sigm(eraw[b * NW + w2 + 1]);
  float v0 = wvec[b * NW + w2], v1 = wvec[b * NW + w2 + 1];
  float w = ww[b * NN + n];
  float2 m = *(const float2*)(mem + ((size_t)(b * NN) + n) * NW + w2);
  float o0 = m.x * (1.f - w * e0) + w * v0;
  float o1 = m.y * (1.f - w * e1) + w * v1;
  *(float2*)(mnew + ((size_t)(b * NN) + n) * NW + w2) = make_float2(o0, o1);
  float ss = o0 * o0 + o1 * o1;
  for (int off = 16; off > 0; off >>= 1) ss += __shfl_xor(ss, off, 32);
  if (lane == 0) mnorm[b * NN + n] = sqrtf(ss);
}

// ------- 4) WMMA streaming pass over link: RowAcc = L*X, ColAcc = L^T*X -------
// Quad-buffered LDS tiles filled by GLOBAL_LOAD_ASYNC_TO_LDS with distance-2
// prefetch; each tile is exactly 3 async instructions, so s_wait_asynccnt 0x3
// guarantees the tile needed next iteration is resident (in-order completion).
__global__ void __launch_bounds__(256) k_link_matvecs(
    const float* __restrict__ link, const float* __restrict__ xbuf,
    float* __restrict__ rowacc, float* __restrict__ pcol) {
  const int b = blockIdx.y;
  const int strip = blockIdx.x;  // 32 strips of 128 rows
  const int ibase = strip * 128;
  const int t = threadIdx.x;
  const int lane = t & 31, wv = t >> 5;
  const int lo = lane & 15, hb = lane >> 4;

  // stride 20 floats (80B): rows 16B-aligned for b128 async writes, and
  // 20*i mod 64 distinct for i=0..15 -> conflict-free fragment reads.
  __shared__ __align__(16) float Lt[4][128][20];
  __shared__ __align__(16) float Xt[4][16][16];
  __shared__ float cred[8 * 256];

  // preload A = X^T fragments for this wave's 16 rows (constant all kernel)
  v2f ax[4];
  {
    int rb = ibase + wv * 16;
    for (int kk = 0; kk < 4; ++kk) {
      int i0 = rb + 4 * kk + hb * 2;
      ax[kk].x = xbuf[((size_t)(b * NN) + i0) * 16 + lo];
      ax[kk].y = xbuf[((size_t)(b * NN) + i0 + 1) * 16 + lo];
    }
  }

  // each thread owns 8 floats (two b128) of the 128x16 L tile
  const int lr = t >> 1;
  const int lc = (t & 1) * 8;
  const uint64_t gL = (uint64_t)(uintptr_t)(link + (size_t)b * NN * NN +
                                            (size_t)(ibase + lr) * NN + lc);
  const uint32_t ldsL = (uint32_t)(uintptr_t)&Lt[0][lr][lc];
  // X tile: one b32 per thread
  const int xjl = t >> 4, xch = t & 15;
  const uint64_t gX =
      (uint64_t)(uintptr_t)(xbuf + ((size_t)(b * NN) + xjl) * 16 + xch);
  const uint32_t ldsX = (uint32_t)(uintptr_t)&Xt[0][xjl][xch];

  auto issue = [&](int tile) {
    uint32_t dl = ldsL + (uint32_t)(tile & 3) * (128u * 20u * 4u);
    uint32_t dx = ldsX + (uint32_t)(tile & 3) * (16u * 16u * 4u);
    uint64_t gl = gL + (uint64_t)tile * 64u;    // +16 floats per tile
    uint64_t gx = gX + (uint64_t)tile * 1024u;  // +16 rows * 16 floats
    asm volatile(
        "global_load_async_to_lds_b128 %0, %2, off\n\t"
        "global_load_async_to_lds_b128 %1, %3, off"
        :: "v"(dl), "v"(dl + 16u), "v"(gl), "v"(gl + 16u) : "memory");
    asm volatile("global_load_async_to_lds_b32 %0, %1, off"
                 :: "v"(dx), "v"(gx) : "memory");
  };

  // prologue: tiles 0 and 1 in flight; tile 0 resident once <=3 outstanding
  issue(0);
  issue(1);
  asm volatile(
      "s_wait_asynccnt 0x3\n\t"
      "s_barrier_signal -1\n\t"
      "s_barrier_wait -1" ::: "memory");

  v8f drow = {0.f, 0.f, 0.f, 0.f, 0.f, 0.f, 0.f, 0.f};
  const int NT = NN / 16;

  for (int jt = 0; jt < NT; ++jt) {
    const int cur = jt & 3;

    if (jt + 2 < NT) issue(jt + 2);  // distance-2 prefetch

    // row pass: drow[i,ch] += L[i, j0+k] * X[j0+k, ch]
#pragma unroll
    for (int kk = 0; kk < 4; ++kk) {
      v2f a, bx;
      int jc = 4 * kk + hb * 2;
      a.x = Lt[cur][wv * 16 + lo][jc];
      a.y = Lt[cur][wv * 16 + lo][jc + 1];
      bx.x = Xt[cur][jc][lo];
      bx.y = Xt[cur][jc + 1][lo];
      drow = __builtin_amdgcn_wmma_f32_16x16x4_f32(false, a, false, bx,
                                                   (short)0, drow, false, false);
    }

    // col pass: dcol[ch,j] += X[i,ch] * L[i,j]   (i over this wave's 16 rows)
    v8f dcol = {0.f, 0.f, 0.f, 0.f, 0.f, 0.f, 0.f, 0.f};
#pragma unroll
    for (int kk = 0; kk < 4; ++kk) {
      v2f bl;
      int ic = wv * 16 + 4 * kk + hb * 2;
      bl.x = Lt[cur][ic][lo];
      bl.y = Lt[cur][ic + 1][lo];
      dcol = __builtin_amdgcn_wmma_f32_16x16x4_f32(false, ax[kk], false, bl,
                                                   (short)0, dcol, false, false);
    }

    // reduce col partial across the 8 waves of this workgroup, store once
#pragma unroll
    for (int v = 0; v < 8; ++v) cred[wv * 256 + v * 32 + lane] = dcol[v];
    __syncthreads();
    {
      float sum = 0.f;
#pragma unroll
      for (int w = 0; w < 8; ++w) sum += cred[w * 256 + t];
      int v = t >> 5, l = t & 31;
      int ch = v + ((l & 16) ? 8 : 0);
      int j = jt * 16 + (l & 15);
      pcol[(((size_t)(b * 32 + strip)) * NN + j) * 16 + ch] = sum;
    }

    // End-of-iteration barrier WITHOUT a storecnt fence: only async-DMA
    // completion (tile jt+1 resident) and LDS quiescence are required.
    if (jt + 2 < NT) {
      asm volatile(
          "s_wait_asynccnt 0x3\n\t"
          "s_wait_dscnt 0x0\n\t"
          "s_barrier_signal -1\n\t"
          "s_barrier_wait -1" ::: "memory");
    } else {
      asm volatile(
          "s_wait_asynccnt 0x0\n\t"
          "s_wait_dscnt 0x0\n\t"
          "s_barrier_signal -1\n\t"
          "s_barrier_wait -1" ::: "memory");
    }
  }

  // write final row accumulators (exclusive ownership, plain stores)
#pragma unroll
  for (int v = 0; v < 8; ++v) {
    int ig = ibase + wv * 16 + v + hb * 8;
    rowacc[((size_t)(b * NN) + ig) * 16 + lo] = drow[v];
  }
}

// ---------------- 5) reduce column partials over the 32 strips ----------------
__global__ void __launch_bounds__(256) k_col_reduce(const float* __restrict__ pcol,
                                                    float* __restrict__ colacc) {
  int g = blockIdx.x * 256 + threadIdx.x;  // B*N*16 threads
  int b = g >> 16;
  int r = g & 65535;
  float s = 0.f;
  const float* base = pcol + ((size_t)b * 32) * NN * 16 + r;
  for (int w = 0; w < 32; ++w) s += base[(size_t)w * NN * 16];
  colacc[(size_t)b * NN * 16 + r] = s;
}

// ---------------- 6) combine into fwd/bwd weights ----------------
__global__ void __launch_bounds__(256) k_combine(
    const float* __restrict__ rowacc, const float* __restrict__ colacc,
    const float* __restrict__ ww, const float* __restrict__ prec,
    const float* __restrict__ rwp, const float* __restrict__ sv,
    const float* __restrict__ tv, float* __restrict__ fwd,
    float* __restrict__ bwd) {
  int g = blockIdx.x * 256 + threadIdx.x;  // B*N threads
  int b = g >> 12, i = g & 4095;
  float wi = ww[b * NN + i], pi = prec[b * NN + i];
  const float* ra = rowacc + ((size_t)(b * NN) + i) * 16;
  const float* ca = colacc + ((size_t)(b * NN) + i) * 16;
  for (int r = 0; r < NR; ++r) {
    float x = rwp[(size_t)(b * NR + r) * NN + i];
    fwd[(size_t)(b * NR + r) * NN + i] =
        (1.f - wi) * ra[r] - ra[r + 4] + wi * (sv[b * NR + r] - pi * x);
    bwd[(size_t)(b * NR + r) * NN + i] =
        (1.f - wi) * ca[r] - ca[r + 4] + pi * (tv[b * NR + r] - wi * x);
  }
}

// ---------------- 7) read-key dot products on memory_new ----------------
__global__ void __launch_bounds__(256) k_read_dots(
    const float* __restrict__ mnew, const float* __restrict__ rkeys,
    float* __restrict__ dots) {
  int t = threadIdx.x, lane = t & 31, wv = t >> 5;
  int row = blockIdx.x * 8 + wv;
  int b = row >> 12, n = row & 4095;
  int w2 = lane * 2;
  float2 m = *(const float2*)(mnew + ((size_t)(b * NN) + n) * NW + w2);
  float acc[NR];
#pragma unroll
  for (int r = 0; r < NR; ++r) {
    float2 k = *(const float2*)(rkeys + (size_t)(b * NR + r) * NW + w2);
    acc[r] = m.x * k.x + m.y * k.y;
  }
#pragma unroll
  for (int r = 0; r < NR; ++r)
    for (int off = 16; off > 0; off >>= 1) acc[r] += __shfl_xor(acc[r], off, 32);
  if (lane == 0) {
#pragma unroll
    for (int r = 0; r < NR; ++r) dots[(size_t)(b * NR + r) * NN + n] = acc[r];
  }
}

// ---------------- 8) read softmax + mode mix -> read weights ----------------
__global__ void __launch_bounds__(256) k_read_weights(
    const float* __restrict__ dots, const float* __restrict__ mnorm,
    const float* __restrict__ rkeys, const float* __restrict__ rsr,
    const float* __restrict__ mraw, const float* __restrict__ fwd,
    const float* __restrict__ bwd, float* __restrict__ rdw) {
  int br = blockIdx.x;
  int b = br >> 2, r = br & 3;
  int tid = threadIdx.x;
  __shared__ float simv[NN];
  __shared__ float red[256];
  float kn2 = 0.f;
  for (int w = 0; w < NW; ++w) { float k = rkeys[(b * NR + r) * NW + w]; kn2 += k * k; }
  float keyn = sqrtf(kn2);
  float beta = 1.f + softplus_(rsr[b * NR + r]);
  float m0 = mraw[(b * NR + r) * 3 + 0];
  float m1 = mraw[(b * NR + r) * 3 + 1];
  float m2 = mraw[(b * NR + r) * 3 + 2];
  float mm = fmaxf(m0, fmaxf(m1, m2));
  float e0 = expf(m0 - mm), e1 = expf(m1 - mm), e2 = expf(m2 - mm);
  float es = e0 + e1 + e2;
  m0 = e0 / es; m1 = e1 / es; m2 = e2 / es;

  float lmax = -3.4e38f;
  for (int p = 0; p < 16; ++p) {
    int n = tid + p * 256;
    float s = beta * dots[(size_t)(b * NR + r) * NN + n] /
              (mnorm[b * NN + n] * keyn + EPSF);
    simv[n] = s;
    lmax = fmaxf(lmax, s);
  }
  red[tid] = lmax;
  __syncthreads();
  for (int off = 128; off > 0; off >>= 1) {
    if (tid < off) red[tid] = fmaxf(red[tid], red[tid + off]);
    __syncthreads();
  }
  float mx = red[0];
  __syncthreads();
  float ps = 0.f;
  for (int p = 0; p < 16; ++p) {
    int n = tid + p * 256;
    float e = expf(simv[n] - mx);
    simv[n] = e;
    ps += e;
  }
  red[tid] = ps;
  __syncthreads();
  for (int off = 128; off > 0; off >>= 1) {
    if (tid < off) red[tid] += red[tid + off];
    __syncthreads();
  }
  float inv = 1.f / red[0];
  for (int p = 0; p < 16; ++p) {
    int n = tid + p * 256;
    size_t o = (size_t)(b * NR + r) * NN + n;
    rdw[o] = m0 * bwd[o] + m1 * simv[n] * inv + m2 * fwd[o];
  }
}

// ---------------- 9) read vectors ----------------
__global__ void __launch_bounds__(256) k_read_vectors(
    const float* __restrict__ mnew, const float* __restrict__ rdw,
    float* __restrict__ out) {
  int b = blockIdx.x;
  int t = threadIdx.x, lane = t & 31, wv = t >> 5;
  int w2 = lane * 2;
  float ax[NR], ay[NR];
  for (int r = 0; r < NR; ++r) { ax[r] = 0.f; ay[r] = 0.f; }
  for (int n = wv; n < NN; n += 8) {
    float2 m = *(const float2*)(mnew + ((size_t)(b * NN) + n) * NW + w2);
#pragma unroll
    for (int r = 0; r < NR; ++r) {
      float rw = rdw[(size_t)(b * NR + r) * NN + n];
      ax[r] += rw * m.x;
      ay[r] += rw * m.y;
    }
  }
  __shared__ float redx[8][32][NR];
  __shared__ float redy[8][32][NR];
  for (int r = 0; r < NR; ++r) { redx[wv][lane][r] = ax[r]; redy[wv][lane][r] = ay[r]; }
  __syncthreads();
  int r = t >> 6, w = t & 63;
  int li = w >> 1, comp = w & 1;
  float s = 0.f;
  for (int q = 0; q < 8; ++q) s += comp ? redy[q][li][r] : redx[q][li][r];
  out[(size_t)(b * NR + r) * NW + w] = s;
}

extern "C" void kernel_launch(void* const* d_in, const int* in_sizes, int n_in,
                              void* d_out, int out_size, void* d_ws, size_t ws_size,
                              hipStream_t stream) {
  (void)in_sizes; (void)n_in; (void)out_size; (void)ws_size;
  const float* memory = (const float*)d_in[0];
  const float* usage  = (const float*)d_in[1];
  const float* link   = (const float*)d_in[2];
  const float* prec   = (const float*)d_in[3];
  const float* rwp    = (const float*)d_in[4];
  const float* wwp    = (const float*)d_in[5];
  const float* wkey   = (const float*)d_in[6];
  const float* wsr    = (const float*)d_in[7];
  const float* eraw   = (const float*)d_in[8];
  const float* wvec   = (const float*)d_in[9];
  const float* fraw   = (const float*)d_in[10];
  const float* agr    = (const float*)d_in[11];
  const float* wgr    = (const float*)d_in[12];
  const float* rkeys  = (const float*)d_in[13];
  const float* rsr    = (const float*)d_in[14];
  const float* mraw   = (const float*)d_in[15];

  float* ws = (float*)d_ws;
  float* cw     = ws + OFF_CW;
  float* ww     = ws + OFF_WW;
  float* sv     = ws + OFF_S;
  float* tv     = ws + OFF_T;
  float* xbuf   = ws + OFF_X;
  float* mnew   = ws + OFF_MN;
  float* mnorm  = ws + OFF_NRM;
  float* rowacc = ws + OFF_RA;
  float* pcol   = ws + OFF_PC;
  float* colacc = ws + OFF_CA;
  float* fwdw   = ws + OFF_FWD;
  float* bwdw   = ws + OFF_BWD;
  float* dotsb  = ws + OFF_DOT;
  float* rdw    = ws + OFF_RDW;

  k_content_write<<<NB, 1024, 0, stream>>>(memory, wkey, wsr, cw);
  k_alloc<<<NB, 1024, 0, stream>>>(usage, rwp, wwp, prec, fraw, agr, wgr, cw,
                                   ww, xbuf, sv, tv);
  k_mem_update<<<(NB * NN) / 8, 256, 0, stream>>>(memory, eraw, wvec, ww, mnew, mnorm);
  k_link_matvecs<<<dim3(32, NB), 256, 0, stream>>>(link, xbuf, rowacc, pcol);
  k_col_reduce<<<(NB * NN * 16) / 256, 256, 0, stream>>>(pcol, colacc);
  k_combine<<<(NB * NN) / 256, 256, 0, stream>>>(rowacc, colacc, ww, prec, rwp,
                                                 sv, tv, fwdw, bwdw);
  k_read_dots<<<(NB * NN) / 8, 256, 0, stream>>>(mnew, rkeys, dotsb);
  k_read_weights<<<NB * NR, 256, 0, stream>>>(dotsb, mnorm, rkeys, rsr, mraw,
                                              fwdw, bwdw, rdw);
  k_read_vectors<<<NB, 256, 0, stream>>>(mnew, rdw, (float*)d_out);
}